// Psf_layer_81003083203167
// MI455X (gfx1250) — compile-verified
//
#include <hip/hip_runtime.h>
#include <hip/hip_bf16.h>
#include <stdint.h>

// ---------------- constants from the reference ----------------
#define PATCH 250
#define WDIM  500
#define NTconst 5
#define NBAND 3
#define NTERMS 64
#define BATCH 4
#define DXC   3.69e-06f
#define DISTC 0.05f
#define DISTCODE 0.047f
#define PAD3C 3000
#define PI_F  3.14159265358979323846f

#define LDBIG 1024   // padded 1000-point DFT domain
#define LDSML 256    // padded 250-point DFT domain
#define NBIG  1000
#define NSML  250

typedef __attribute__((ext_vector_type(2))) float v2f;
typedef __attribute__((ext_vector_type(8))) float v8f;
typedef __attribute__((ext_vector_type(4))) unsigned int u32x4;
typedef __attribute__((ext_vector_type(8))) int i32x8;
typedef __attribute__((ext_vector_type(4))) int i32x4;

#if defined(__has_builtin)
#if __has_builtin(__builtin_amdgcn_tensor_load_to_lds)
#define HAVE_TDM 1
#else
#define HAVE_TDM 0
#endif
#else
#define HAVE_TDM 0
#endif

#if defined(__has_builtin)
#if __has_builtin(__builtin_amdgcn_s_wait_tensorcnt)
#define HAVE_TWAIT 1
#else
#define HAVE_TWAIT 0
#endif
#else
#define HAVE_TWAIT 0
#endif

// Chunk of K staged in LDS per iteration; LDS row stride = KT+1 (via TDM padding)
#define KT 32
#define LDS_ROW (KT + 1)
#define PLANE_FLTS (32 * LDS_ROW)     // 32 rows * 33 floats = 1056

// The TDM builtin carries no pointer operand, so the compiler cannot see that it
// writes LDS. LDS_PUBLISH captures the array address inside an opaque asm, so
// alias analysis must assume external writes to it; LDS_WRITE_FENCE then acts
// as an ordering point, forcing the As[] reads to be real ds_loads per chunk.
#define LDS_PUBLISH(p)    asm volatile("" :: "v"(p) : "memory")
#define LDS_WRITE_FENCE() asm volatile("" ::: "memory")

// Issue a TDM 2D-tile load: rows x cols f32 tile from gsrc (row stride ld floats)
// into LDS at byte offset ldsByteOff, padding 1 DWORD after every 32 DWORDs
// (=> LDS row stride 33 floats). D# packing per CDNA5 ISA ch.8 / CDNA5 ISA.
__device__ __forceinline__ void tdm_load_tile(const float* gsrc, float* ldsDst,
                                              unsigned ldsByteOff, int ld,
                                              int rows, int cols) {
#if HAVE_TDM
  unsigned long long ga = (unsigned long long)(uintptr_t)gsrc;
  u32x4 g0;
  g0.x = 1u;                                              // count=1 (valid user D#)
  g0.y = ldsByteOff;                                      // lds_addr
  g0.z = (unsigned)(ga & 0xFFFFFFFFu);                    // global_addr[31:0]
  g0.w = (unsigned)((ga >> 32) & 0x01FFFFFFu) | (2u << 30); // global_addr[56:32] | type=2
  unsigned d0 = (unsigned)ld, d1 = (unsigned)ld;
  unsigned long long s0 = (unsigned long long)(unsigned)ld;
  i32x8 g1;
  g1[0] = (int)((2u << 16) | (1u << 20) | (4u << 22));    // data_size=4B, pad_enable, pad_interval=32 DW, pad_amount=1 DW
  g1[1] = (int)((d0 & 0xFFFFu) << 16);                    // tensor_dim0[15:0]
  g1[2] = (int)(((d0 >> 16) & 0xFFFFu) | ((d1 & 0xFFFFu) << 16));
  g1[3] = (int)(((d1 >> 16) & 0xFFFFu) | ((unsigned)cols << 16)); // tile_dim0
  g1[4] = (int)rows;                                      // tile_dim1 (tile_dim2=0)
  g1[5] = (int)(s0 & 0xFFFFFFFFu);                        // tensor_dim0_stride lo
  g1[6] = (int)((s0 >> 32) & 0xFFFFu);                    // stride hi | dim1_stride=0
  g1[7] = 0;
  i32x4 z4 = {0, 0, 0, 0};
  i32x8 z8 = {0, 0, 0, 0, 0, 0, 0, 0};
  __builtin_amdgcn_tensor_load_to_lds(g0, g1, z4, z4, z8, 0);
  (void)ldsDst;
#else
  // fallback: cooperative copy by the issuing wave (32 lanes)
  int lane = threadIdx.x & 31;
  for (int idx = lane; idx < rows * cols; idx += 32) {
    int r = idx / cols, c = idx % cols;
    ldsDst[r * LDS_ROW + c] = gsrc[(size_t)r * ld + c];
  }
#endif
}

// ---------------- complex GEMM via WMMA f32 16x16x4 + TDM-staged A ----------------
// C = A * B (complex), matrices ld x ld row-major f32 planes.
// Block = 128 threads = 4 waves. Block computes a 32(M) x 128(N) C macro-tile:
// every wave computes a 32x32 patch (2x2 WMMA tiles, 16 v8f accumulators).
// All waves share the same 32-row A panel -> staged once per K-chunk via TDM.
__global__ __launch_bounds__(128) void k_cgemm(
    float* __restrict__ Cre, float* __restrict__ Cim,
    const float* __restrict__ Are, const float* __restrict__ Aim,
    const float* __restrict__ Bre, const float* __restrict__ Bim,
    int ld)
{
  __shared__ float As[2 * PLANE_FLTS];  // [re | im], rows padded to 33 floats
  float* AsRe = As;
  float* AsIm = As + PLANE_FLTS;

  // Capture the LDS address so the compiler must assume the TDM engine
  // (invisible to alias analysis) writes it.
  LDS_PUBLISH(As);

  const int lane = threadIdx.x & 31;
  const int wave = threadIdx.x >> 5;
  const int half = lane >> 4;          // selects K pair {0,1} vs {2,3}
  const int r16  = lane & 15;
  const int m0 = blockIdx.y * 32;                       // 2 M tiles: m0, m0+16
  const int n0 = blockIdx.x * 128 + wave * 32;          // 2 N tiles: n0, n0+16

  v8f acc[2][2][4];
  #pragma unroll
  for (int mt = 0; mt < 2; ++mt)
    #pragma unroll
    for (int nt = 0; nt < 2; ++nt)
      #pragma unroll
      for (int p = 0; p < 4; ++p)
        acc[mt][nt][p] = (v8f){};

  for (int kc = 0; kc < ld; kc += KT) {
    LDS_WRITE_FENCE();                 // order previous reads before DMA overwrite
    if (wave == 0) {
      tdm_load_tile(Are + (size_t)m0 * ld + kc, AsRe, 0u,              ld, 32, KT);
      tdm_load_tile(Aim + (size_t)m0 * ld + kc, AsIm, 4u * PLANE_FLTS, ld, 32, KT);
#if HAVE_TDM && HAVE_TWAIT
      __builtin_amdgcn_s_wait_tensorcnt((short)0);
#endif
    }
    __syncthreads();
    LDS_WRITE_FENCE();                 // As[] now holds DMA-written data

    #pragma unroll
    for (int kk = 0; kk < KT; kk += 4) {
      const int ka = kk + 2 * half;
      v2f ar[2], ai[2];
      #pragma unroll
      for (int mt = 0; mt < 2; ++mt) {
        const int ro = (mt * 16 + r16) * LDS_ROW + ka;
        ar[mt].x = AsRe[ro]; ar[mt].y = AsRe[ro + 1];
        ai[mt].x = AsIm[ro]; ai[mt].y = AsIm[ro + 1];
      }
      v2f br[2], bi[2];
      #pragma unroll
      for (int nt = 0; nt < 2; ++nt) {
        const float* pr = Bre + (size_t)(kc + ka) * ld + (n0 + nt * 16 + r16);
        const float* pi = Bim + (size_t)(kc + ka) * ld + (n0 + nt * 16 + r16);
        br[nt].x = pr[0]; br[nt].y = pr[ld];
        bi[nt].x = pi[0]; bi[nt].y = pi[ld];
      }
      #pragma unroll
      for (int mt = 0; mt < 2; ++mt)
        #pragma unroll
        for (int nt = 0; nt < 2; ++nt) {
          acc[mt][nt][0] = __builtin_amdgcn_wmma_f32_16x16x4_f32(false, ar[mt], false, br[nt], (short)0, acc[mt][nt][0], false, false);
          acc[mt][nt][1] = __builtin_amdgcn_wmma_f32_16x16x4_f32(false, ai[mt], false, bi[nt], (short)0, acc[mt][nt][1], false, false);
          acc[mt][nt][2] = __builtin_amdgcn_wmma_f32_16x16x4_f32(false, ar[mt], false, bi[nt], (short)0, acc[mt][nt][2], false, false);
          acc[mt][nt][3] = __builtin_amdgcn_wmma_f32_16x16x4_f32(false, ai[mt], false, br[nt], (short)0, acc[mt][nt][3], false, false);
        }
    }
    __syncthreads();
  }

  // C/D layout: VGPR r -> M = tile_m0 + half*8 + r, N = tile_n0 + (lane&15)
  #pragma unroll
  for (int mt = 0; mt < 2; ++mt)
    #pragma unroll
    for (int nt = 0; nt < 2; ++nt)
      #pragma unroll
      for (int r = 0; r < 8; ++r) {
        size_t idx = (size_t)(m0 + mt * 16 + half * 8 + r) * ld + (n0 + nt * 16 + r16);
        Cre[idx] = acc[mt][nt][0][r] - acc[mt][nt][1][r];
        Cim[idx] = acc[mt][nt][2][r] + acc[mt][nt][3][r];
      }
}

// ---------------- DFT matrix builder (zero-padded to ld) ----------------
__global__ void k_build_dft(float* __restrict__ cosm, float* __restrict__ sinm,
                            int n, int ld, float sinSign, float scale)
{
  int k = blockIdx.x * blockDim.x + threadIdx.x;
  int j = blockIdx.y * blockDim.y + threadIdx.y;
  if (j >= ld || k >= ld) return;
  float c = 0.f, s = 0.f;
  if (j < n && k < n) {
    long long m = ((long long)j * (long long)k) % (long long)n;
    float th = (float)(2.0 * 3.14159265358979323846 * (double)m / (double)n);
    c = cosf(th) * scale;
    s = sinSign * sinf(th) * scale;
  }
  cosm[(size_t)j * ld + k] = c;
  sinm[(size_t)j * ld + k] = s;
}

// ---------------- small elementwise kernels ----------------
__global__ void k_zero(float* __restrict__ p, int n) {
  int i = blockIdx.x * blockDim.x + threadIdx.x;
  if (i < n) p[i] = 0.f;
}

__global__ void k_height(float* __restrict__ h, const float* __restrict__ zc,
                         const float* __restrict__ zv) {
  int idx = blockIdx.x * blockDim.x + threadIdx.x;
  if (idx >= WDIM * WDIM) return;
  float acc = 0.f;
  #pragma unroll 4
  for (int t = 0; t < NTERMS; ++t) acc += zc[t] * zv[(size_t)t * WDIM * WDIM + idx];
  h[idx] = acc;
}

__global__ void k_mask(float* __restrict__ mask, const float* __restrict__ mp,
                       const float* __restrict__ fr, const float* __restrict__ fg,
                       const float* __restrict__ fc, const float* __restrict__ fb) {
  int idx = blockIdx.x * blockDim.x + threadIdx.x;
  if (idx >= WDIM * WDIM * NBAND) return;
  int b = idx % NBAND;
  int ij = idx / NBAND;
  int j = ij % WDIM, i = ij / WDIM;
  int r = i % (2 * NTconst); if (r >= NTconst) r = 2 * NTconst - 1 - r;
  int c = j % (2 * NTconst); if (c >= NTconst) c = 2 * NTconst - 1 - c;
  int o = r * NTconst + c;
  float v = fr[b] * mp[0 * 25 + o] + fg[b] * mp[1 * 25 + o] +
            fc[b] * mp[2 * 25 + o] + fb[b] * mp[3 * 25 + o];
  mask[idx] = v;
}

__global__ void k_field(float* __restrict__ re, float* __restrict__ im,
                        const float* __restrict__ h, const float* __restrict__ wls,
                        const float* __restrict__ ridx, int band) {
  int idx = blockIdx.x * blockDim.x + threadIdx.x;
  if (idx >= LDBIG * LDBIG) return;
  int i = idx >> 10, j = idx & (LDBIG - 1);
  float vr = 0.f, vi = 0.f;
  if (i >= 250 && i < 750 && j >= 250 && j < 750) {
    int a = i - 250, b = j - 250;
    float dy = (float)(a - 250), dx = (float)(b - 250);
    if (dy * dy + dx * dx < 249.f * 249.f) {
      float ph = 2.f * PI_F / wls[band] * ridx[band] * h[a * WDIM + b];
      vr = cosf(ph); vi = sinf(ph);
    }
  }
  re[idx] = vr; im[idx] = vi;
}

__global__ void k_phase(float* __restrict__ re, float* __restrict__ im,
                        const float* __restrict__ wls, int band, float dist) {
  int idx = blockIdx.x * blockDim.x + threadIdx.x;
  if (idx >= NBIG * NBIG) return;
  int j = idx / NBIG, c = idx % NBIG;
  const float inv = 1.0f / ((float)NBIG * DXC);
  float fj = (float)((j < NBIG / 2) ? j : j - NBIG) * inv;
  float fc_ = (float)((c < NBIG / 2) ? c : c - NBIG) * inv;
  float ph = -PI_F * dist * wls[band] * (fj * fj + fc_ * fc_);
  float cr = cosf(ph), ci = sinf(ph);
  size_t o = (size_t)j * LDBIG + c;
  float r0 = re[o], i0 = im[o];
  re[o] = r0 * cr - i0 * ci;
  im[o] = r0 * ci + i0 * cr;
}

__global__ void k_crop(float* __restrict__ ur, float* __restrict__ ui,
                       const float* __restrict__ br, const float* __restrict__ bi) {
  int idx = blockIdx.x * blockDim.x + threadIdx.x;
  if (idx >= WDIM * WDIM) return;
  int i = idx / WDIM, j = idx % WDIM;
  size_t o = (size_t)(i + 250) * LDBIG + (j + 250);
  ur[idx] = br[o]; ui[idx] = bi[o];
}

__global__ void k_tilefield(float* __restrict__ re, float* __restrict__ im,
                            const float* __restrict__ ur, const float* __restrict__ ui,
                            const float* __restrict__ mask, int band, int s0, int s1) {
  int idx = blockIdx.x * blockDim.x + threadIdx.x;
  if (idx >= LDBIG * LDBIG) return;
  int i = idx >> 10, j = idx & (LDBIG - 1);
  float vr = 0.f, vi = 0.f;
  if (i >= 250 && i < 750 && j >= 250 && j < 750) {
    int a = i - 250, b = j - 250;
    int ma = ((a - s0) % WDIM + WDIM) % WDIM;
    int mb = ((b - s1) % WDIM + WDIM) % WDIM;
    float m = mask[(ma * WDIM + mb) * NBAND + band];
    vr = ur[a * WDIM + b] * m;
    vi = ui[a * WDIM + b] * m;
  }
  re[idx] = vr; im[idx] = vi;
}

__global__ void k_psf(float* __restrict__ psf, const float* __restrict__ br,
                      const float* __restrict__ bi) {
  int idx = blockIdx.x * blockDim.x + threadIdx.x;
  if (idx >= PATCH * PATCH) return;
  int p = idx / PATCH, q = idx % PATCH;
  int r0 = 26 * p - PAD3C, c0 = 26 * q - PAD3C;
  float s = 0.f;
  for (int rr = 0; rr < 26; ++rr) {
    int r = r0 + rr;
    if ((unsigned)r >= (unsigned)WDIM) continue;
    for (int cc = 0; cc < 26; ++cc) {
      int c = c0 + cc;
      if ((unsigned)c >= (unsigned)WDIM) continue;
      size_t o = (size_t)(r + 250) * LDBIG + (c + 250);
      float x = br[o], y = bi[o];
      s += x * x + y * y;
    }
  }
  psf[idx] = s * (1.0f / 676.0f);
}

__global__ void k_reduce(const float* __restrict__ v, int n, float* __restrict__ out) {
  __shared__ float sm[256];
  float a = 0.f;
  for (int i = threadIdx.x; i < n; i += 256) a += v[i];
  sm[threadIdx.x] = a;
  __syncthreads();
  for (int s = 128; s > 0; s >>= 1) {
    if ((int)threadIdx.x < s) sm[threadIdx.x] += sm[threadIdx.x + s];
    __syncthreads();
  }
  if (threadIdx.x == 0) *out = sm[0];
}

__global__ void k_psfshift(float* __restrict__ dst, const float* __restrict__ psf,
                           const float* __restrict__ sum) {
  int idx = blockIdx.x * blockDim.x + threadIdx.x;
  if (idx >= LDSML * LDSML) return;
  int i = idx >> 8, j = idx & (LDSML - 1);
  float v = 0.f;
  if (i < NSML && j < NSML) {
    v = psf[((i + 125) % NSML) * NSML + ((j + 125) % NSML)] / *sum;
  }
  dst[idx] = v;
}

__global__ void k_convx(float* __restrict__ xt, const float* __restrict__ inputs,
                        int band, int k, int bi) {
  int idx = blockIdx.x * blockDim.x + threadIdx.x;
  if (idx >= LDSML * LDSML) return;
  int i = idx >> 8, j = idx & (LDSML - 1);
  float v = 0.f;
  if (i < NSML && j < NSML && (i % NTconst) == (k / NTconst) && (j % NTconst) == (k % NTconst)) {
    v = inputs[(((size_t)bi * PATCH + i) * PATCH + j) * NBAND + band];
  }
  xt[idx] = v;
}

__global__ void k_cmulacc(float* __restrict__ ar, float* __restrict__ ai,
                          const float* __restrict__ xr, const float* __restrict__ xi,
                          const float* __restrict__ pr, const float* __restrict__ pi) {
  int idx = blockIdx.x * blockDim.x + threadIdx.x;
  if (idx >= LDSML * LDSML) return;
  float a = xr[idx], b = xi[idx], c = pr[idx], d = pi[idx];
  ar[idx] += a * c - b * d;
  ai[idx] += a * d + b * c;
}

__global__ void k_accout(float* __restrict__ out, const float* __restrict__ sre,
                         const float* __restrict__ fr, const float* __restrict__ fg,
                         const float* __restrict__ fb, int band, int bi) {
  int idx = blockIdx.x * blockDim.x + threadIdx.x;
  if (idx >= PATCH * PATCH) return;
  int i = idx / PATCH, j = idx % PATCH;
  float s = sre[i * LDSML + j];
  size_t base = (((size_t)bi * PATCH + i) * PATCH + j) * 3;
  out[base + 0] += fr[band] * s;
  out[base + 1] += fg[band] * s;
  out[base + 2] += fb[band] * s;
}

// ---------------- host orchestration ----------------
static inline int nblk(int n) { return (n + 255) / 256; }

extern "C" void kernel_launch(void* const* d_in, const int* in_sizes, int n_in,
                              void* d_out, int out_size, void* d_ws, size_t ws_size,
                              hipStream_t stream) {
  const float* inputs = (const float*)d_in[0];
  const float* zc     = (const float*)d_in[1];
  const float* zv     = (const float*)d_in[2];
  const float* mp     = (const float*)d_in[3];
  const float* fr     = (const float*)d_in[4];
  const float* fg     = (const float*)d_in[5];
  const float* fc     = (const float*)d_in[6];
  const float* fb     = (const float*)d_in[7];
  const float* ridx   = (const float*)d_in[8];
  const float* wls    = (const float*)d_in[9];
  float* out = (float*)d_out;

  float* W0 = (float*)d_ws;
  size_t off = 0;
  auto alloc = [&](size_t n) { float* p = W0 + off; off += n; return p; };
  const size_t D1 = (size_t)LDBIG * LDBIG, S1 = (size_t)LDSML * LDSML;
  float* Dfc = alloc(D1); float* Dfs = alloc(D1);
  float* Dic = alloc(D1); float* Dis = alloc(D1);
  float* dfc = alloc(S1); float* dfs = alloc(S1);
  float* dic = alloc(S1); float* dis = alloc(S1);
  float* Hgt = alloc((size_t)WDIM * WDIM);
  float* Msk = alloc((size_t)WDIM * WDIM * NBAND);
  float* Xre = alloc(D1); float* Xim = alloc(D1);
  float* Tre = alloc(D1); float* Tim = alloc(D1);
  float* Yre = alloc(D1); float* Yim = alloc(D1);
  float* U1r = alloc((size_t)WDIM * WDIM); float* U1i = alloc((size_t)WDIM * WDIM);
  float* Psf = alloc((size_t)PATCH * PATCH); float* PsfSum = alloc(64);
  float* PsfSh = alloc(S1);
  float* Z256 = alloc(S1);
  float* FpR = alloc(25 * S1); float* FpI = alloc(25 * S1);
  float* Xt  = alloc(S1);
  float* t2r = alloc(S1); float* t2i = alloc(S1);
  float* fxr = alloc(S1); float* fxi = alloc(S1);
  float* accr = alloc(S1); float* acci = alloc(S1);
  float* sr  = alloc(S1); float* si = alloc(S1);

  auto cgemm = [&](float* cr, float* ci, const float* arr, const float* aii,
                   const float* brr, const float* bii, int ld) {
    dim3 g(ld / 128, ld / 32), b(128);
    k_cgemm<<<g, b, 0, stream>>>(cr, ci, arr, aii, brr, bii, ld);
  };

  {
    dim3 b(16, 16);
    dim3 gB(LDBIG / 16, LDBIG / 16), gS(LDSML / 16, LDSML / 16);
    k_build_dft<<<gB, b, 0, stream>>>(Dfc, Dfs, NBIG, LDBIG, -1.f, 1.f);
    k_build_dft<<<gB, b, 0, stream>>>(Dic, Dis, NBIG, LDBIG, +1.f, 1.f / (float)NBIG);
    k_build_dft<<<gS, b, 0, stream>>>(dfc, dfs, NSML, LDSML, -1.f, 1.f);
    k_build_dft<<<gS, b, 0, stream>>>(dic, dis, NSML, LDSML, +1.f, 1.f / (float)NSML);
  }
  k_height<<<nblk(WDIM * WDIM), 256, 0, stream>>>(Hgt, zc, zv);
  k_mask<<<nblk(WDIM * WDIM * NBAND), 256, 0, stream>>>(Msk, mp, fr, fg, fc, fb);
  k_zero<<<nblk((int)S1), 256, 0, stream>>>(Z256, (int)S1);
  k_zero<<<nblk(out_size), 256, 0, stream>>>(out, out_size);

  const int mov[NTconst] = {0, -1, -1, -2, -2};
  const int nBigEl = LDBIG * LDBIG;

  auto prop = [&](float dist, int band) {
    cgemm(Tre, Tim, Dfc, Dfs, Xre, Xim, LDBIG);
    cgemm(Yre, Yim, Tre, Tim, Dfc, Dfs, LDBIG);
    k_phase<<<nblk(NBIG * NBIG), 256, 0, stream>>>(Yre, Yim, wls, band, dist);
    cgemm(Tre, Tim, Dic, Dis, Yre, Yim, LDBIG);
    cgemm(Xre, Xim, Tre, Tim, Dic, Dis, LDBIG);
  };

  for (int band = 0; band < NBAND; ++band) {
    k_field<<<nblk(nBigEl), 256, 0, stream>>>(Xre, Xim, Hgt, wls, ridx, band);
    prop(DISTCODE, band);
    k_crop<<<nblk(WDIM * WDIM), 256, 0, stream>>>(U1r, U1i, Xre, Xim);

    for (int k = 0; k < NTconst * NTconst; ++k) {
      int s0 = mov[k / NTconst], s1 = mov[k % NTconst];
      k_tilefield<<<nblk(nBigEl), 256, 0, stream>>>(Xre, Xim, U1r, U1i, Msk, band, s0, s1);
      prop(-DISTCODE, band);
      prop(DISTC, band);
      k_psf<<<nblk(PATCH * PATCH), 256, 0, stream>>>(Psf, Xre, Xim);
      k_reduce<<<1, 256, 0, stream>>>(Psf, PATCH * PATCH, PsfSum);
      k_psfshift<<<nblk((int)S1), 256, 0, stream>>>(PsfSh, Psf, PsfSum);
      cgemm(t2r, t2i, dfc, dfs, PsfSh, Z256, LDSML);
      cgemm(FpR + (size_t)k * S1, FpI + (size_t)k * S1, t2r, t2i, dfc, dfs, LDSML);
    }

    for (int bi = 0; bi < BATCH; ++bi) {
      k_zero<<<nblk((int)S1), 256, 0, stream>>>(accr, (int)S1);
      k_zero<<<nblk((int)S1), 256, 0, stream>>>(acci, (int)S1);
      for (int k = 0; k < NTconst * NTconst; ++k) {
        k_convx<<<nblk((int)S1), 256, 0, stream>>>(Xt, inputs, band, k, bi);
        cgemm(t2r, t2i, dfc, dfs, Xt, Z256, LDSML);
        cgemm(fxr, fxi, t2r, t2i, dfc, dfs, LDSML);
        k_cmulacc<<<nblk((int)S1), 256, 0, stream>>>(accr, acci, fxr, fxi,
                                                     FpR + (size_t)k * S1,
                                                     FpI + (size_t)k * S1);
      }
      cgemm(t2r, t2i, dic, dis, accr, acci, LDSML);
      cgemm(sr, si, t2r, t2i, dic, dis, LDSML);
      k_accout<<<nblk(PATCH * PATCH), 256, 0, stream>>>(out, sr, fr, fg, fb, band, bi);
    }
  }
  (void)in_sizes; (void)n_in; (void)ws_size;
}